// LocalNeighborhood_37649683317414
// MI455X (gfx1250) — compile-verified
//
#include <hip/hip_runtime.h>
#include <stdint.h>

// Problem constants (match reference): B=4, L=4096, D=128, KMAX=16
#define B_ 4
#define L_ 4096
#define D_ 128
#define K_ 16
#define ITEMS_PER_WAVE 2           // double-buffered TDM pipeline per wave
#define WAVES_PER_BLOCK 4

typedef float f32x4 __attribute__((ext_vector_type(4)));
typedef unsigned int u32x4 __attribute__((ext_vector_type(4)));
typedef int i32x8 __attribute__((ext_vector_type(8)));
typedef int i32x4 __attribute__((ext_vector_type(4)));

#if defined(__AMDGCN__) && __has_builtin(__builtin_amdgcn_tensor_load_to_lds) && \
    __has_builtin(__builtin_amdgcn_tensor_store_from_lds) &&                     \
    __has_builtin(__builtin_amdgcn_s_wait_tensorcnt)
#define USE_TDM 1
#else
#define USE_TDM 0
#endif

#if USE_TDM
__device__ __forceinline__ unsigned rfl_u32(unsigned x) {
  return (unsigned)__builtin_amdgcn_readfirstlane((int)x);
}
#endif

// Two-pointer K=16 nearest-neighbor search on sorted 1-D coords.
// All 32 lanes run it redundantly (uniform), so the neighbor list lands in
// registers. Also writes the |d_index| output row (lanes 0..15, coalesced).
__device__ __forceinline__ void knn16_and_idist(const int* __restrict__ c, int i,
                                                int lane,
                                                float* __restrict__ idist_row,
                                                int nIdx[K_]) {
  const float ci = (float)c[i];
  float nD2[K_];
  float nAbs[K_];

  int l = i - 1;
  int r = i;  // self is first right-candidate (d2 = 0)
  float d2l = 3.0e38f, abl = 0.f;
  float d2r = 3.0e38f, abr = 0.f;
  if (l >= 0) { float d = (float)c[l] - ci; abl = __builtin_fabsf(d); d2l = d * d; }
  if (r < L_) { float d = (float)c[r] - ci; abr = __builtin_fabsf(d); d2r = d * d; }

#pragma unroll
  for (int k = 0; k < K_; ++k) {
    // tie (d2l == d2r) -> take left: left index is always smaller (top_k rule)
    const bool takeLeft = (l >= 0) && (!(r < L_) || (d2l <= d2r));
    if (takeLeft) {
      nIdx[k] = l; nD2[k] = d2l; nAbs[k] = abl;
      --l;
      d2l = 3.0e38f;
      if (l >= 0) { float d = (float)c[l] - ci; abl = __builtin_fabsf(d); d2l = d * d; }
    } else {
      nIdx[k] = r; nD2[k] = d2r; nAbs[k] = abr;
      ++r;
      d2r = 3.0e38f;
      if (r < L_) { float d = (float)c[r] - ci; abr = __builtin_fabsf(d); d2r = d * d; }
    }
  }

  // Fix tie ordering among equal d2 (top_k: ascending index). List is already
  // sorted by d2; tiny unrolled bubble sort on (d2, idx) in registers.
#pragma unroll
  for (int a = 0; a < K_ - 1; ++a) {
#pragma unroll
    for (int t = 0; t < K_ - 1 - a; ++t) {
      const bool sw = (nD2[t + 1] < nD2[t]) ||
                      (nD2[t + 1] == nD2[t] && nIdx[t + 1] < nIdx[t]);
      if (sw) {
        int   ti = nIdx[t]; nIdx[t] = nIdx[t + 1]; nIdx[t + 1] = ti;
        float td = nD2[t];  nD2[t]  = nD2[t + 1];  nD2[t + 1]  = td;
        float ta = nAbs[t]; nAbs[t] = nAbs[t + 1]; nAbs[t + 1] = ta;
      }
    }
  }

  float myd = 0.f;
#pragma unroll
  for (int t = 0; t < K_; ++t)
    if (lane == t) myd = nAbs[t];
  if (lane < K_)
    idist_row[lane] = myd;
}

#if USE_TDM
// Issue a TDM gather-mode load: 16 rows of 128 f32 from attr[b] into LDS.
__device__ __forceinline__ void tdm_gather_load(unsigned lds, const float* src_base,
                                                const int nIdx[K_]) {
  unsigned pk[8];
#pragma unroll
  for (int t = 0; t < 8; ++t)
    pk[t] = rfl_u32(((unsigned)nIdx[2 * t] & 0xFFFFu) |
                    (((unsigned)nIdx[2 * t + 1] & 0xFFFFu) << 16));

  const uint64_t ga = (uint64_t)(uintptr_t)src_base;
  u32x4 g0;
  g0.x = 1u | (1u << 31);                      // count=1, gather_mode=1, 16-bit idx
  g0.y = lds;                                  // lds_addr
  g0.z = rfl_u32((unsigned)ga);                // global_addr[31:0]
  g0.w = (rfl_u32((unsigned)(ga >> 32)) & 0x01FFFFFFu) | (2u << 30);  // [56:32], type=2
  i32x8 g1 = { (int)(2u << 16),                // data_size=4B, wg_mask=0
               (int)(D_ << 16),                // tensor_dim0 = 128 (lo16)
               (int)(L_ << 16),                // tensor_dim1 = 4096 (lo16)
               (int)(D_ << 16),                // tile_dim0 = 128 (row width)
               K_,                             // tile_dim1 = #valid indices = 16
               D_,                             // tensor_dim0_stride = 128
               0, 0 };
  i32x4 g2 = { (int)pk[0], (int)pk[1], (int)pk[2], (int)pk[3] };
  i32x4 g3 = { (int)pk[4], (int)pk[5], (int)pk[6], (int)pk[7] };

#if __clang_major__ >= 23
  i32x8 z8 = { 0, 0, 0, 0, 0, 0, 0, 0 };
  __builtin_amdgcn_tensor_load_to_lds(g0, g1, g2, g3, z8, 0);
#else
  __builtin_amdgcn_tensor_load_to_lds(g0, g1, g2, g3, 0);
#endif
}

// Issue a TDM store of the contiguous 16x128 f32 tile from LDS to out.
__device__ __forceinline__ void tdm_store_tile(unsigned lds, float* dst) {
  const uint64_t ga = (uint64_t)(uintptr_t)dst;
  u32x4 s0;
  s0.x = 1u;                                   // count=1, no gather
  s0.y = lds;
  s0.z = rfl_u32((unsigned)ga);
  s0.w = (rfl_u32((unsigned)(ga >> 32)) & 0x01FFFFFFu) | (2u << 30);
  i32x8 s1 = { (int)(2u << 16),                // data_size=4B
               (int)((K_ * D_) << 16),         // tensor_dim0 = 2048 (lo16)
               (int)(1u << 16),                // tensor_dim1 = 1
               (int)((K_ * D_) << 16),         // tile_dim0 = 2048
               0,                              // tile_dim1 unused (1-D tile)
               K_ * D_,                        // tensor_dim0_stride = 2048
               0, 0 };
  i32x4 z4 = { 0, 0, 0, 0 };
#if __clang_major__ >= 23
  i32x8 z8 = { 0, 0, 0, 0, 0, 0, 0, 0 };
  __builtin_amdgcn_tensor_store_from_lds(s0, s1, z4, z4, z8, 0);
#else
  __builtin_amdgcn_tensor_store_from_lds(s0, s1, z4, z4, 0);
#endif
}
#endif  // USE_TDM

// One wave (32 lanes) per ITEMS_PER_WAVE consecutive (b, i) items.
// TDM path double-buffers two 8 KB LDS tiles and pipelines:
//   L0, L1, wait<=1 (L0 done), S0, wait<=1 (L1 done), S1
// exploiting per-wave in-order TENSORcnt retirement.
__global__ __launch_bounds__(32 * WAVES_PER_BLOCK) void
LocalNeighborhood_kernel(const int* __restrict__ fidx,   // [B, L, 1] int32 (sorted along L)
                         const float* __restrict__ attr, // [B, L, D] f32
                         float* __restrict__ out) {      // [B*L*K] idist ++ [B*L*K*D] attr_nb
  const int lane = threadIdx.x & 31;
  const int wv   = threadIdx.x >> 5;
  const int pair = blockIdx.x * WAVES_PER_BLOCK + wv;    // 0 .. B*L/2-1
  const int b    = pair >> 11;                           // / (L_/ITEMS_PER_WAVE)
  const int i0   = (pair & ((L_ / ITEMS_PER_WAVE) - 1)) * ITEMS_PER_WAVE;

  const int* c = fidx + b * L_;
  const float* attr_b = attr + (size_t)b * L_ * D_;
  float* const out_attr_base = out + (size_t)B_ * L_ * K_;

  int nIdx0[K_], nIdx1[K_];
  knn16_and_idist(c, i0 + 0, lane, out + ((size_t)b * L_ + i0 + 0) * K_, nIdx0);
  knn16_and_idist(c, i0 + 1, lane, out + ((size_t)b * L_ + i0 + 1) * K_, nIdx1);

  float* const dst0 = out_attr_base + ((size_t)b * L_ + i0 + 0) * (K_ * D_);
  float* const dst1 = out_attr_base + ((size_t)b * L_ + i0 + 1) * (K_ * D_);

#if USE_TDM
  __shared__ float tile[WAVES_PER_BLOCK][ITEMS_PER_WAVE][K_ * D_];  // 2 x 8 KB / wave

  const unsigned lds0 = rfl_u32((unsigned)(uintptr_t)&tile[wv][0][0]);
  const unsigned lds1 = rfl_u32((unsigned)(uintptr_t)&tile[wv][1][0]);

  tdm_gather_load(lds0, attr_b, nIdx0);   // TENSORcnt = 1
  tdm_gather_load(lds1, attr_b, nIdx1);   // TENSORcnt = 2
  __builtin_amdgcn_s_wait_tensorcnt(1);   // load0 retired (in-order)
  tdm_store_tile(lds0, dst0);             // TENSORcnt = 2 (load1, store0)
  __builtin_amdgcn_s_wait_tensorcnt(1);   // load1 retired (in-order)
  tdm_store_tile(lds1, dst1);
  // S_ENDPGM performs an implicit wait-idle (TENSORcnt -> 0)

#else  // !USE_TDM ----------------------------------------------------------
  // Fallback: float4 gather loads + non-temporal streaming stores
  const f32x4* __restrict__ a4 = (const f32x4*)attr_b;
  f32x4* __restrict__ o40 = (f32x4*)dst0;
  f32x4* __restrict__ o41 = (f32x4*)dst1;
#pragma unroll
  for (int k = 0; k < K_; ++k) {
    const f32x4 v0 = a4[(size_t)nIdx0[k] * (D_ / 4) + lane];
    __builtin_nontemporal_store(v0, &o40[k * (D_ / 4) + lane]);
  }
#pragma unroll
  for (int k = 0; k < K_; ++k) {
    const f32x4 v1 = a4[(size_t)nIdx1[k] * (D_ / 4) + lane];
    __builtin_nontemporal_store(v1, &o41[k * (D_ / 4) + lane]);
  }
#endif
}

extern "C" void kernel_launch(void* const* d_in, const int* in_sizes, int n_in,
                              void* d_out, int out_size, void* d_ws, size_t ws_size,
                              hipStream_t stream) {
  (void)in_sizes; (void)n_in; (void)d_ws; (void)ws_size; (void)out_size;
  const int*   fidx = (const int*)d_in[0];    // first_index [B, L, 1] int32
  const float* attr = (const float*)d_in[1];  // attr        [B, L, D] f32
  float*       out  = (float*)d_out;          // idist ++ attr_nb (f32)

  const int total_waves = B_ * L_ / ITEMS_PER_WAVE;     // one wave per 2 items
  dim3 grid(total_waves / WAVES_PER_BLOCK);             // 2048 blocks
  dim3 block(32 * WAVES_PER_BLOCK);                     // 4 waves (wave32)
  LocalNeighborhood_kernel<<<grid, block, 0, stream>>>(fidx, attr, out);
}